// ComplexMultiHeadAttention_1365799600288
// MI455X (gfx1250) — compile-verified
//
#include <hip/hip_runtime.h>
#include <hip/hip_bf16.h>
#include <math.h>

typedef __attribute__((ext_vector_type(2))) float v2f;
typedef __attribute__((ext_vector_type(8))) float v8f;

#define F_DIM 1024
#define BATCH 4
#define SEQ   1024
#define HEADS 16
#define HDIM  64
#define MTOT  (BATCH * SEQ)
#define SMS   1026   // LDS row stride (floats): even (8B-aligned v2f) and bank-spread

// D = A(16x4 f32) * B(4x16 f32) + C(16x16 f32), full-precision CDNA5 WMMA
__device__ __forceinline__ v8f wmma4(v2f a, v2f b, v8f c) {
  return __builtin_amdgcn_wmma_f32_16x16x4_f32(
      /*neg_a=*/false, a, /*neg_b=*/false, b,
      /*c_mod=*/(short)0, c, /*reuse_a=*/false, /*reuse_b=*/false);
}

// ---------------------------------------------------------------------------
// Complex GEMM: Yr = Xr*Wr - Xi*Wi ; Yi = Xr*Wi + Xi*Wr   (M x 1024 x 1024)
// Optional residual add on the epilogue.
// Block: 256 threads = 8 waves (4 M x 2 N), block tile 128x64, wave tile 32x32.
// ---------------------------------------------------------------------------
__global__ __launch_bounds__(256)
void cgemm_kernel(const float* __restrict__ Xr, const float* __restrict__ Xi,
                  const float* __restrict__ Wr, const float* __restrict__ Wi,
                  float* __restrict__ Yr, float* __restrict__ Yi,
                  const float* __restrict__ Rr, const float* __restrict__ Ri,
                  int hasRes) {
  const int lane = threadIdx.x & 31;
  const int w    = threadIdx.x >> 5;
  const int wy   = w >> 1;          // 0..3 (M direction)
  const int wx   = w & 1;           // 0..1 (N direction)
  const int m0   = blockIdx.x * 128 + wy * 32;
  const int n0   = blockIdx.y * 64  + wx * 32;
  const int lm   = lane & 15;
  const int c0   = (lane < 16) ? 0 : 2;   // K sub-offset per ISA A/B layout

  v8f acc_r[2][2], acc_i[2][2];
  for (int mt = 0; mt < 2; ++mt)
    for (int nt = 0; nt < 2; ++nt) { acc_r[mt][nt] = (v8f){}; acc_i[mt][nt] = (v8f){}; }

  size_t aoff[2], boff[2];
  for (int mt = 0; mt < 2; ++mt)
    aoff[mt] = (size_t)(m0 + mt * 16 + lm) * F_DIM + c0;
  for (int nt = 0; nt < 2; ++nt)
    boff[nt] = (size_t)c0 * F_DIM + (size_t)(n0 + nt * 16 + lm);

  for (int k = 0; k < F_DIM; k += 4) {
    v2f ar[2], ai[2];
    for (int mt = 0; mt < 2; ++mt) {
      ar[mt] = *(const v2f*)(Xr + aoff[mt] + k);
      ai[mt] = *(const v2f*)(Xi + aoff[mt] + k);
    }
    v2f br[2], bi[2], nbi[2];
    for (int nt = 0; nt < 2; ++nt) {
      size_t o = boff[nt] + (size_t)k * F_DIM;
      br[nt].x = Wr[o];  br[nt].y = Wr[o + F_DIM];
      bi[nt].x = Wi[o];  bi[nt].y = Wi[o + F_DIM];
      nbi[nt].x = -bi[nt].x; nbi[nt].y = -bi[nt].y;
    }
    for (int mt = 0; mt < 2; ++mt)
      for (int nt = 0; nt < 2; ++nt) {
        acc_r[mt][nt] = wmma4(ar[mt], br[nt],  acc_r[mt][nt]);
        acc_r[mt][nt] = wmma4(ai[mt], nbi[nt], acc_r[mt][nt]);
        acc_i[mt][nt] = wmma4(ar[mt], bi[nt],  acc_i[mt][nt]);
        acc_i[mt][nt] = wmma4(ai[mt], br[nt],  acc_i[mt][nt]);
      }
  }

  for (int mt = 0; mt < 2; ++mt)
    for (int nt = 0; nt < 2; ++nt)
      for (int r = 0; r < 8; ++r) {
        int row = m0 + mt * 16 + r + ((lane < 16) ? 0 : 8);
        int col = n0 + nt * 16 + lm;
        size_t o = (size_t)row * F_DIM + col;
        float vr = acc_r[mt][nt][r];
        float vi = acc_i[mt][nt][r];
        if (hasRes) { vr += Rr[o]; vi += Ri[o]; }
        Yr[o] = vr; Yi[o] = vi;
      }
}

// ---------------------------------------------------------------------------
// Attention: one block per (b, h, 64-query strip).
//  stage 1: complex scores over D=64, |s| -> LDS [64][SMS]
//  stage 2: row softmax (wave-per-row, wave32 shuffles), write attn to global
//  stage 3: attn @ v (A from LDS) -> O heads buffer [B, L, F]
// ---------------------------------------------------------------------------
__global__ __launch_bounds__(256)
void attn_kernel(const float* __restrict__ Qr, const float* __restrict__ Qi,
                 const float* __restrict__ Kr, const float* __restrict__ Ki,
                 const float* __restrict__ Vr, const float* __restrict__ Vi,
                 float* __restrict__ Or_, float* __restrict__ Oi_,
                 float* __restrict__ attn_out) {
  extern __shared__ float smem[];   // 64 * SMS floats (~257 KB of the 320 KB WGP LDS)
  const int lane = threadIdx.x & 31;
  const int w    = threadIdx.x >> 5;
  const int b    = blockIdx.y >> 4;
  const int h    = blockIdx.y & 15;
  const int q0   = blockIdx.x * 64;
  const int hoff = h * HDIM;
  const int lm   = lane & 15;
  const int c0   = (lane < 16) ? 0 : 2;
  const size_t baseBL = (size_t)b * SEQ * F_DIM;

  // ---- stage 1: scores -> magnitudes in LDS ----
  const int wy = w >> 1;            // q 16-row strip 0..3
  const int wx = w & 1;             // k half (512 cols)
  const int qs = q0 + wy * 16;

  v2f qrf[16], qif[16], nqrf[16];   // all of D=64 held in registers
  {
    const float inv_temp = 0.125f;  // 1/sqrt(D)
    size_t qoff = baseBL + (size_t)(qs + lm) * F_DIM + hoff + c0;
    for (int s = 0; s < 16; ++s) {
      v2f a  = *(const v2f*)(Qr + qoff + s * 4);
      v2f bq = *(const v2f*)(Qi + qoff + s * 4);
      a.x *= inv_temp; a.y *= inv_temp;
      bq.x *= inv_temp; bq.y *= inv_temp;
      qrf[s] = a; qif[s] = bq;
      nqrf[s].x = -a.x; nqrf[s].y = -a.y;
    }
  }

  for (int kt = 0; kt < 32; ++kt) {
    const int kcol = wx * 512 + kt * 16;
    v8f cre = (v8f){}; v8f cim = (v8f){};
    size_t kb = baseBL + (size_t)(kcol + lm) * F_DIM + hoff + c0;
    for (int s = 0; s < 16; ++s) {
      v2f bkr = *(const v2f*)(Kr + kb + s * 4);
      v2f bki = *(const v2f*)(Ki + kb + s * 4);
      cre = wmma4(qrf[s],  bkr, cre);   // qr.kr
      cre = wmma4(qif[s],  bki, cre);   // + qi.ki
      cim = wmma4(qif[s],  bkr, cim);   // qi.kr
      cim = wmma4(nqrf[s], bki, cim);   // - qr.ki
    }
    for (int r = 0; r < 8; ++r) {
      int row = wy * 16 + r + ((lane < 16) ? 0 : 8);
      int col = kcol + lm;
      float re = cre[r], im = cim[r];
      smem[row * SMS + col] = sqrtf(re * re + im * im);
    }
  }
  __syncthreads();

  // ---- stage 2: softmax, one wave per row batch ----
  for (int rr = 0; rr < 8; ++rr) {
    const int row = w * 8 + rr;
    float* srow = smem + row * SMS;
    float mx = -3.0e38f;
    for (int j = lane; j < SEQ; j += 32) mx = fmaxf(mx, srow[j]);
    for (int off = 16; off > 0; off >>= 1) mx = fmaxf(mx, __shfl_xor(mx, off, 32));
    float sum = 0.f;
    for (int j = lane; j < SEQ; j += 32) {
      float e = __expf(srow[j] - mx);
      srow[j] = e;
      sum += e;
    }
    for (int off = 16; off > 0; off >>= 1) sum += __shfl_xor(sum, off, 32);
    const float inv = 1.0f / sum;
    size_t abase = ((size_t)blockIdx.y * SEQ + (q0 + row)) * SEQ;
    for (int j = lane; j < SEQ; j += 32) {
      float a = srow[j] * inv;
      srow[j] = a;
      attn_out[abase + j] = a;
    }
  }
  __syncthreads();

  // ---- stage 3: O = attn @ V (A fragments straight from LDS) ----
  const int qt = w >> 1;            // q tile 0..3
  const int dh = (w & 1) * 32;      // d half
  v8f o_r[2], o_i[2];
  o_r[0] = (v8f){}; o_r[1] = (v8f){}; o_i[0] = (v8f){}; o_i[1] = (v8f){};
  size_t aoffL = (size_t)(qt * 16 + lm) * SMS + c0;
  size_t vbase = baseBL + (size_t)c0 * F_DIM + hoff + dh + lm;
  for (int k = 0; k < SEQ; k += 4) {
    v2f a = *(const v2f*)(smem + aoffL + k);
    size_t vo = vbase + (size_t)k * F_DIM;
    v2f br0, br1, bi0, bi1;
    br0.x = Vr[vo];       br0.y = Vr[vo + F_DIM];
    br1.x = Vr[vo + 16];  br1.y = Vr[vo + F_DIM + 16];
    bi0.x = Vi[vo];       bi0.y = Vi[vo + F_DIM];
    bi1.x = Vi[vo + 16];  bi1.y = Vi[vo + F_DIM + 16];
    o_r[0] = wmma4(a, br0, o_r[0]);
    o_r[1] = wmma4(a, br1, o_r[1]);
    o_i[0] = wmma4(a, bi0, o_i[0]);
    o_i[1] = wmma4(a, bi1, o_i[1]);
  }
  for (int nt = 0; nt < 2; ++nt)
    for (int r = 0; r < 8; ++r) {
      int row = q0 + qt * 16 + r + ((lane < 16) ? 0 : 8);
      int col = hoff + dh + nt * 16 + lm;
      size_t o = baseBL + (size_t)row * F_DIM + col;
      Or_[o] = o_r[nt][r];
      Oi_[o] = o_i[nt][r];
    }
}

// ---------------------------------------------------------------------------
// LayerNorm over F=1024, one block per row.
// ---------------------------------------------------------------------------
__global__ __launch_bounds__(256)
void ln_kernel(const float* __restrict__ X, const float* __restrict__ g,
               const float* __restrict__ be, float* __restrict__ out) {
  __shared__ float red[256];
  const int row = blockIdx.x;
  const int t = threadIdx.x;
  const float* x = X + (size_t)row * F_DIM;
  float vals[4];
  float s = 0.f;
  for (int j = 0; j < 4; ++j) { vals[j] = x[t + 256 * j]; s += vals[j]; }
  red[t] = s;
  __syncthreads();
  for (int off = 128; off > 0; off >>= 1) {
    if (t < off) red[t] += red[t + off];
    __syncthreads();
  }
  const float mu = red[0] * (1.0f / F_DIM);
  __syncthreads();
  float v = 0.f;
  for (int j = 0; j < 4; ++j) { float d = vals[j] - mu; v += d * d; }
  red[t] = v;
  __syncthreads();
  for (int off = 128; off > 0; off >>= 1) {
    if (t < off) red[t] += red[t + off];
    __syncthreads();
  }
  const float inv = rsqrtf(red[0] * (1.0f / F_DIM) + 1e-6f);
  float* o = out + (size_t)row * F_DIM;
  for (int j = 0; j < 4; ++j) {
    int c = t + 256 * j;
    o[c] = g[c] * (vals[j] - mu) * inv + be[c];
  }
}

// ---------------------------------------------------------------------------
extern "C" void kernel_launch(void* const* d_in, const int* in_sizes, int n_in,
                              void* d_out, int out_size, void* d_ws, size_t ws_size,
                              hipStream_t stream) {
  const float* q_r  = (const float*)d_in[0];
  const float* q_i  = (const float*)d_in[1];
  const float* k_r  = (const float*)d_in[2];
  const float* k_i  = (const float*)d_in[3];
  const float* v_r  = (const float*)d_in[4];
  const float* v_i  = (const float*)d_in[5];
  const float* Wq_r = (const float*)d_in[6];
  const float* Wq_i = (const float*)d_in[7];
  const float* Wk_r = (const float*)d_in[8];
  const float* Wk_i = (const float*)d_in[9];
  const float* Wv_r = (const float*)d_in[10];
  const float* Wv_i = (const float*)d_in[11];
  const float* Wf_r = (const float*)d_in[12];
  const float* Wf_i = (const float*)d_in[13];
  const float* g_r  = (const float*)d_in[14];
  const float* b_r  = (const float*)d_in[15];
  const float* g_i  = (const float*)d_in[16];
  const float* b_i  = (const float*)d_in[17];

  float* ws = (float*)d_ws;
  const size_t SZ = (size_t)MTOT * F_DIM;           // 4M floats
  float* Qr = ws + 0 * SZ;  float* Qi = ws + 1 * SZ;
  float* Kr = ws + 2 * SZ;  float* Ki = ws + 3 * SZ;
  float* Vr = ws + 4 * SZ;  float* Vi = ws + 5 * SZ;
  float* Oh_r = ws + 6 * SZ; float* Oh_i = ws + 7 * SZ;
  float* Pr = Kr;  // K no longer needed post-attention: reuse for pre-LN buffers
  float* Pi = Ki;

  float* out   = (float*)d_out;
  float* out_r = out;
  float* out_i = out + SZ;
  float* attn  = out + 2 * SZ;                      // [B,H,L,L]

  dim3 gg(MTOT / 128, F_DIM / 64), bb(256);
  cgemm_kernel<<<gg, bb, 0, stream>>>(q_r, q_i, Wq_r, Wq_i, Qr, Qi, nullptr, nullptr, 0);
  cgemm_kernel<<<gg, bb, 0, stream>>>(k_r, k_i, Wk_r, Wk_i, Kr, Ki, nullptr, nullptr, 0);
  cgemm_kernel<<<gg, bb, 0, stream>>>(v_r, v_i, Wv_r, Wv_i, Vr, Vi, nullptr, nullptr, 0);

  attn_kernel<<<dim3(SEQ / 64, BATCH * HEADS), 256, 64 * SMS * sizeof(float), stream>>>(
      Qr, Qi, Kr, Ki, Vr, Vi, Oh_r, Oh_i, attn);

  cgemm_kernel<<<gg, bb, 0, stream>>>(Oh_r, Oh_i, Wf_r, Wf_i, Pr, Pi, q_r, q_i, 1);

  ln_kernel<<<MTOT, 256, 0, stream>>>(Pr, g_r, b_r, out_r);
  ln_kernel<<<MTOT, 256, 0, stream>>>(Pi, g_i, b_i, out_i);
}